// SensorGCN_20950850470229
// MI455X (gfx1250) — compile-verified
//
#include <hip/hip_runtime.h>
#include <hip/hip_bf16.h>
#include <math.h>

typedef __attribute__((ext_vector_type(2))) float v2f;
typedef __attribute__((ext_vector_type(8))) float v8f;

#define THREADS 256

// ---------------- utility ----------------
__global__ void k_zero(float* __restrict__ p, long long n) {
  long long i = (long long)blockIdx.x * blockDim.x + threadIdx.x;
  long long stride = (long long)gridDim.x * blockDim.x;
  for (; i < n; i += stride) p[i] = 0.0f;
}

// ---------------- gcn_norm ----------------
// deg = segment_sum(ew, col) over E edges + 1.0 per node (self loop)
__global__ void k_deg(const int* __restrict__ ei, const float* __restrict__ ew,
                      float* __restrict__ deg, int E, int N) {
  long long i = (long long)blockIdx.x * blockDim.x + threadIdx.x;
  if (i >= (long long)E + N) return;
  if (i < E) {
    atomicAdd(&deg[ei[(long long)E + i]], ew[i]);   // col = ei[1][e]
  } else {
    atomicAdd(&deg[i - E], 1.0f);                   // self loop weight 1
  }
}

// dis = deg>0 ? rsqrt(max(deg,1e-20)) : 0  (in place)
__global__ void k_dis(float* __restrict__ deg, int N) {
  int n = blockIdx.x * blockDim.x + threadIdx.x;
  if (n >= N) return;
  float d = deg[n];
  deg[n] = d > 0.0f ? rsqrtf(fmaxf(d, 1e-20f)) : 0.0f;
}

// norm[e]   = dis[row]*ew*dis[col]   (e < E)
// norm[E+n] = dis[n]*dis[n]          (self loops)
__global__ void k_norm(const int* __restrict__ ei, const float* __restrict__ ew,
                       const float* __restrict__ dis, float* __restrict__ norm,
                       int E, int N) {
  long long i = (long long)blockIdx.x * blockDim.x + threadIdx.x;
  if (i >= (long long)E + N) return;
  if (i < E) {
    norm[i] = dis[ei[i]] * ew[i] * dis[ei[(long long)E + i]];
  } else {
    float d = dis[i - E];
    norm[i] = d * d;
  }
}

// ---------------- dense GEMM via V_WMMA_F32_16X16X4_F32 ----------------
// out[nrows x 64] = A[nrows x kin] @ W[kin x 64]   (row-major)
// block = 128 threads (4 waves); each wave owns one 16x16 tile;
// blockIdx.x selects the 16-row band, wave id selects the 16-col band.
__global__ void k_gemm_wmma(const float* __restrict__ A, const float* __restrict__ W,
                            float* __restrict__ out, int nrows, int kin) {
  const int lane  = threadIdx.x & 31;
  const int wave  = threadIdx.x >> 5;
  const int mBase = blockIdx.x * 16;
  const int nBase = wave * 16;
  const int half  = lane >> 4;   // 0: lanes 0-15, 1: lanes 16-31
  const int l     = lane & 15;

  int arowIdx = mBase + l;
  if (arowIdx >= nrows) arowIdx = nrows - 1;  // clamp (keeps EXEC full for WMMA)

  const float* ap = A + (size_t)arowIdx * kin + 2 * half;       // K = k0 + 2*half
  const float* wp = W + (size_t)(2 * half) * 64 + nBase + l;    // row k0+2*half, col nBase+l

  v8f c = {};
  for (int k0 = 0; k0 < kin; k0 += 4) {
    v2f a; a.x = ap[0];  a.y = ap[1];     // K = k0+2h, k0+2h+1 for row M=l
    v2f b; b.x = wp[0];  b.y = wp[64];    // rows k0+2h, k0+2h+1 at col nBase+l
    c = __builtin_amdgcn_wmma_f32_16x16x4_f32(
        /*neg_a=*/false, a, /*neg_b=*/false, b,
        /*c_mod=*/(short)0, c, /*reuse_a=*/false, /*reuse_b=*/false);
    ap += 4;
    wp += 4 * 64;
  }

  // D layout: VGPR i -> M = mBase + 8*half + i, N = nBase + l
  const int orow0 = mBase + 8 * half;
#pragma unroll
  for (int i = 0; i < 8; ++i) {
    int r = orow0 + i;
    if (r < nrows) out[(size_t)r * 64 + nBase + l] = c[i];
  }
}

// ---------------- message passing: acc[col] += norm * t[row] ----------------
// one thread per (edge-or-loop, feature); 64 consecutive lanes share one item
__global__ void k_scatter(const float* __restrict__ t, const int* __restrict__ ei,
                          const float* __restrict__ norm, float* __restrict__ acc,
                          int E, int N) {
  long long idx = (long long)blockIdx.x * blockDim.x + threadIdx.x;
  long long total = ((long long)E + N) * 64;
  if (idx >= total) return;
  int f = (int)(idx & 63);
  long long item = idx >> 6;
  int r, c;
  if (item < E) {
    r = ei[item];
    c = ei[(long long)E + item];
  } else {
    r = c = (int)(item - E);   // self loop
  }
  float v = norm[item] * t[(size_t)r * 64 + f];
  atomicAdd(&acc[(size_t)c * 64 + f], v);
}

// acc[n*64+f] = (acc + b[f]), optional relu
__global__ void k_bias_act(float* __restrict__ acc, const float* __restrict__ b,
                           long long N, int relu) {
  long long idx = (long long)blockIdx.x * blockDim.x + threadIdx.x;
  if (idx >= N * 64) return;
  float v = acc[idx] + b[idx & 63];
  acc[idx] = relu ? fmaxf(v, 0.0f) : v;
}

// ---------------- pooling + head ----------------
__global__ void k_pool(const float* __restrict__ h, const int* __restrict__ batch,
                       float* __restrict__ sums, float* __restrict__ cnt, int N) {
  long long idx = (long long)blockIdx.x * blockDim.x + threadIdx.x;
  if (idx >= (long long)N * 64) return;
  int f = (int)(idx & 63);
  int n = (int)(idx >> 6);
  int g = batch[n];
  atomicAdd(&sums[(size_t)g * 64 + f], h[idx]);
  if (f == 0) atomicAdd(&cnt[g], 1.0f);
}

__global__ void k_head(const float* __restrict__ sums, const float* __restrict__ cnt,
                       const float* __restrict__ Wl, const float* __restrict__ bl,
                       float* __restrict__ out, int G) {
  int g = blockIdx.x * blockDim.x + threadIdx.x;
  if (g >= G) return;
  float inv = 1.0f / fmaxf(cnt[g], 1.0f);
  float l0 = bl[0], l1 = bl[1], l2 = bl[2];
  const float* s = sums + (size_t)g * 64;
#pragma unroll 8
  for (int j = 0; j < 64; ++j) {
    float p = s[j] * inv;
    l0 += p * Wl[j * 3 + 0];
    l1 += p * Wl[j * 3 + 1];
    l2 += p * Wl[j * 3 + 2];
  }
  float m = fmaxf(l0, fmaxf(l1, l2));
  float e0 = expf(l0 - m), e1 = expf(l1 - m), e2 = expf(l2 - m);
  float inv_s = 1.0f / (e0 + e1 + e2);
  out[(size_t)g * 3 + 0] = e0 * inv_s;
  out[(size_t)g * 3 + 1] = e1 * inv_s;
  out[(size_t)g * 3 + 2] = e2 * inv_s;
}

static inline int nblk(long long n) { return (int)((n + THREADS - 1) / THREADS); }

extern "C" void kernel_launch(void* const* d_in, const int* in_sizes, int n_in,
                              void* d_out, int out_size, void* d_ws, size_t ws_size,
                              hipStream_t stream) {
  const float* x  = (const float*)d_in[0];
  const int*   ei = (const int*)d_in[1];
  const float* ew = (const float*)d_in[2];
  const int*   batch = (const int*)d_in[3];
  const float* W1 = (const float*)d_in[4];
  const float* b1 = (const float*)d_in[5];
  const float* W2 = (const float*)d_in[6];
  const float* b2 = (const float*)d_in[7];
  const float* W3 = (const float*)d_in[8];
  const float* b3 = (const float*)d_in[9];
  const float* Wl = (const float*)d_in[10];
  const float* bl = (const float*)d_in[11];
  float* out = (float*)d_out;

  const int N = in_sizes[0] / 32;   // F_IN = 32
  const int E = in_sizes[2];        // edge_weight has E elements
  const int G = out_size / 3;       // C = 3

  // workspace layout (floats)
  float* ws   = (float*)d_ws;
  float* deg  = ws;                                  // N   (becomes dis in place)
  float* norm = deg + N;                             // E+N
  float* T    = norm + (size_t)E + N;                // N*64 (GEMM output)
  float* Fa   = T  + (size_t)N * 64;                 // N*64
  float* Fb   = Fa + (size_t)N * 64;                 // N*64
  float* sums = Fb + (size_t)N * 64;                 // G*64
  float* cnt  = sums + (size_t)G * 64;               // G

  const long long NE   = (long long)E + N;
  const long long NF   = (long long)N * 64;
  const long long MSG  = NE * 64;
  const int gemmBlocks = (N + 15) / 16;

  // --- gcn_norm (shared across all 3 layers) ---
  k_zero<<<2048, THREADS, 0, stream>>>(deg, N);
  k_deg <<<nblk(NE), THREADS, 0, stream>>>(ei, ew, deg, E, N);
  k_dis <<<nblk(N),  THREADS, 0, stream>>>(deg, N);
  k_norm<<<nblk(NE), THREADS, 0, stream>>>(ei, ew, deg, norm, E, N);

  // --- layer 1: relu(agg(x@W1) + b1) -> Fa ---
  k_gemm_wmma<<<gemmBlocks, 128, 0, stream>>>(x, W1, T, N, 32);
  k_zero    <<<4096, THREADS, 0, stream>>>(Fa, NF);
  k_scatter <<<nblk(MSG), THREADS, 0, stream>>>(T, ei, norm, Fa, E, N);
  k_bias_act<<<nblk(NF),  THREADS, 0, stream>>>(Fa, b1, N, 1);

  // --- layer 2: relu(agg(Fa@W2) + b2) -> Fb ---
  k_gemm_wmma<<<gemmBlocks, 128, 0, stream>>>(Fa, W2, T, N, 64);
  k_zero    <<<4096, THREADS, 0, stream>>>(Fb, NF);
  k_scatter <<<nblk(MSG), THREADS, 0, stream>>>(T, ei, norm, Fb, E, N);
  k_bias_act<<<nblk(NF),  THREADS, 0, stream>>>(Fb, b2, N, 1);

  // --- layer 3: agg(Fb@W3) + b3 -> Fa (no relu) ---
  k_gemm_wmma<<<gemmBlocks, 128, 0, stream>>>(Fb, W3, T, N, 64);
  k_zero    <<<4096, THREADS, 0, stream>>>(Fa, NF);
  k_scatter <<<nblk(MSG), THREADS, 0, stream>>>(T, ei, norm, Fa, E, N);
  k_bias_act<<<nblk(NF),  THREADS, 0, stream>>>(Fa, b3, N, 0);

  // --- global mean pool + linear + softmax ---
  k_zero<<<64, THREADS, 0, stream>>>(sums, (long long)G * 64 + G);
  k_pool<<<nblk(NF), THREADS, 0, stream>>>(Fa, batch, sums, cnt, N);
  k_head<<<nblk(G), THREADS, 0, stream>>>(sums, cnt, Wl, bl, out, G);
}